// Classifier_6717328851414
// MI455X (gfx1250) — compile-verified
//
#include <hip/hip_runtime.h>

typedef __attribute__((ext_vector_type(2))) float v2f;
typedef __attribute__((ext_vector_type(8))) float v8f;

#define NITER   150
#define PLANE   1296                 // 36*36 padded plane (pad=2 each side)
#define PPAIR   (2 * PLANE)          // one interleaved channel-pair block (floats)
// element (ic, off) lives at base + (ic>>1)*PPAIR + off*2 + (ic&1)
#define IN8_OFF 0                    // 4 pairs: (z0,z1)(z2,z3)(z4,x0)(x1,x2)
#define HB_OFF  (8 * PLANE)          // 4 pairs: (h0,h1)(h2,h3)(h4,h5)(0,0)
#define WA1_OFF (16 * PLANE)         // 50 chunks * 32 lanes * v2f = 3200 floats
#define WA2_OFF (16 * PLANE + 3200)  // 3200 floats
#define B1_OFF  (16 * PLANE + 6400)
#define B2_OFF  (B1_OFF + 16)
#define LOG_OFF (B2_OFF + 16)
#define LDS_FLOATS (LOG_OFF + 16)    // 27184 floats = 108736 bytes

// One 5x5/pad2 conv over 8 (padded, pair-interleaved) input planes.
// D(16x16) = W(16x4) x patch(4x16): M = out-channel (padded to 16),
// N = 16 consecutive pixels of one row, K = 4 input channels per chunk,
// 2 halves x 25 taps = 50 WMMAs per tile, in two independent 25-deep chains.
// MODE 0: in8 -> h = leaky(conv+b1) into hb. MODE 1: hb -> z = 0.5z+0.5*leaky(conv+b2).
template <int MODE>
__device__ __forceinline__ void conv_pass(float* lds, const float* wA, const float* bias)
{
    const int tid  = threadIdx.x;
    const int lane = tid & 31;
    const int wv   = tid >> 5;       // 8 waves
    const int n    = lane & 15;      // N index within tile
    const int hi   = lane >> 4;      // half-wave: K rows +2

    float* inp  = lds + (MODE == 0 ? IN8_OFF : HB_OFF);
    float* outp = lds + (MODE == 0 ? HB_OFF  : IN8_OFF);

    for (int tile = wv; tile < 64; tile += 8) {     // 8 tiles per wave, uniform
        const int pix = tile * 16 + n;
        const int row = pix >> 5;
        const int col = pix & 31;
        // lane's patch base: channel pair selected by hi, interleaved offset *2
        const float* pt = inp + hi * PPAIR + (row * 36 + col) * 2;

        v8f acc0 = {};
        v8f acc1 = {};
#pragma unroll
        for (int pos = 0; pos < 25; ++pos) {
            const int dy = pos / 5, dx = pos % 5;
            {   // half 0: input channels 0..3
                const int c = pos * 2;
                v2f a = *(const v2f*)(wA + c * 64 + lane * 2);
                v2f b = *(const v2f*)(pt + (dy * 36 + dx) * 2);
                acc0 = __builtin_amdgcn_wmma_f32_16x16x4_f32(
                    false, a, false, b, (short)0, acc0, false, false);
            }
            {   // half 1: input channels 4..7 (+2 pair blocks)
                const int c = pos * 2 + 1;
                v2f a = *(const v2f*)(wA + c * 64 + lane * 2);
                v2f b = *(const v2f*)(pt + 2 * PPAIR + (dy * 36 + dx) * 2);
                acc1 = __builtin_amdgcn_wmma_f32_16x16x4_f32(
                    false, a, false, b, (short)0, acc1, false, false);
            }
        }

        // D layout: lanes 0-15 hold M=v (oc), N=lane; lanes 16-31 hold oc>=8 (unused)
        if (hi == 0) {
            const int off = (row + 2) * 36 + (col + 2);
            if (MODE == 0) {
#pragma unroll
                for (int v = 0; v < 6; ++v) {
                    float val = acc0[v] + acc1[v] + bias[v];
                    val = val > 0.f ? val : 0.01f * val;
                    outp[(v >> 1) * PPAIR + off * 2 + (v & 1)] = val;
                }
            } else {
#pragma unroll
                for (int v = 0; v < 5; ++v) {
                    float val = acc0[v] + acc1[v] + bias[v];
                    val = val > 0.f ? val : 0.01f * val;
                    float* zp = outp + (v >> 1) * PPAIR + off * 2 + (v & 1);
                    *zp = 0.5f * (*zp) + 0.5f * val;
                }
            }
        }
    }
}

__global__ __launch_bounds__(256) void deq_kernel(
    const float* __restrict__ image,
    const float* __restrict__ w1, const float* __restrict__ b1,
    const float* __restrict__ w2, const float* __restrict__ b2,
    const float* __restrict__ wh, const float* __restrict__ bh,
    float* __restrict__ out)
{
    extern __shared__ float lds[];
    const int tid = threadIdx.x;
    const int img = blockIdx.x;

    // Zero all 16 data planes (z starts at 0; pad borders & spare slots stay 0).
    for (int i = tid; i < 16 * PLANE; i += 256) lds[i] = 0.f;

    // Repack conv weights into the per-lane WMMA A layout:
    // chunk c: pos = c>>1 (tap), half = c&1; lane l: oc = l&15, icbase = half*4 + (l>>4)*2.
    for (int idx = tid; idx < 1600; idx += 256) {
        const int c = idx >> 5, l = idx & 31;
        const int oc = l & 15, hi = l >> 4;
        const int pos = c >> 1, half = c & 1;
        const int ic0 = half * 4 + hi * 2;
        float x = 0.f, y = 0.f;
        if (oc < 6) {                                // w1: (6,8,5,5) OIHW
            x = w1[(oc * 8 + ic0) * 25 + pos];
            y = w1[(oc * 8 + ic0 + 1) * 25 + pos];
        }
        lds[WA1_OFF + idx * 2]     = x;
        lds[WA1_OFF + idx * 2 + 1] = y;
        x = 0.f; y = 0.f;
        if (oc < 5) {                                // w2: (5,6,5,5), ic padded 6->8
            if (ic0     < 6) x = w2[(oc * 6 + ic0) * 25 + pos];
            if (ic0 + 1 < 6) y = w2[(oc * 6 + ic0 + 1) * 25 + pos];
        }
        lds[WA2_OFF + idx * 2]     = x;
        lds[WA2_OFF + idx * 2 + 1] = y;
    }
    if (tid < 16) {
        lds[B1_OFF + tid]  = (tid < 6) ? b1[tid] : 0.f;
        lds[B2_OFF + tid]  = (tid < 5) ? b2[tid] : 0.f;
        lds[LOG_OFF + tid] = 0.f;
    }
    __syncthreads();

    // Image interior into x slots (global channels 5..7, pair-interleaved).
    for (int i = tid; i < 3 * 1024; i += 256) {
        const int ch = i >> 10, p = i & 1023;
        const int r = p >> 5, cc = p & 31;
        const int off = (r + 2) * 36 + (cc + 2);
        const int ic = 5 + ch;
        lds[IN8_OFF + (ic >> 1) * PPAIR + off * 2 + (ic & 1)] = image[img * 3072 + i];
    }
    // Warm L2 with head weights (global_prefetch_b8); one byte per cacheline stride.
    for (int i = tid * 32; i < 10 * 5 * 1024; i += 256 * 32)
        __builtin_prefetch(wh + i, 0, 1);
    __syncthreads();

    // 150 damped fixed-point iterations, entirely LDS-resident.
    for (int it = 0; it < NITER; ++it) {
        conv_pass<0>(lds, lds + WA1_OFF, lds + B1_OFF);
        __syncthreads();
        conv_pass<1>(lds, lds + WA2_OFF, lds + B2_OFF);
        __syncthreads();
    }

    // Class head: logits[oc] = sum_{ch,pix} z * wh + bh  (wh is L2-resident).
    float part[10];
#pragma unroll
    for (int oc = 0; oc < 10; ++oc) part[oc] = 0.f;
    for (int p = tid; p < 1024; p += 256) {
        const int r = p >> 5, cc = p & 31;
        const int off = (r + 2) * 36 + (cc + 2);
        float zv[5];
#pragma unroll
        for (int ch = 0; ch < 5; ++ch)
            zv[ch] = lds[IN8_OFF + (ch >> 1) * PPAIR + off * 2 + (ch & 1)];
#pragma unroll
        for (int oc = 0; oc < 10; ++oc) {
            float s = 0.f;
#pragma unroll
            for (int ch = 0; ch < 5; ++ch)
                s += zv[ch] * wh[(oc * 5 + ch) * 1024 + p];
            part[oc] += s;
        }
    }
#pragma unroll
    for (int oc = 0; oc < 10; ++oc) atomicAdd(&lds[LOG_OFF + oc], part[oc]);
    __syncthreads();
    if (tid < 10) out[img * 10 + tid] = lds[LOG_OFF + tid] + bh[tid];
}

extern "C" void kernel_launch(void* const* d_in, const int* in_sizes, int n_in,
                              void* d_out, int out_size, void* d_ws, size_t ws_size,
                              hipStream_t stream) {
    const float* image = (const float*)d_in[0];
    const float* w1    = (const float*)d_in[1];
    const float* b1    = (const float*)d_in[2];
    const float* w2    = (const float*)d_in[3];
    const float* b2    = (const float*)d_in[4];
    const float* wh    = (const float*)d_in[5];
    const float* bh    = (const float*)d_in[6];
    float* out = (float*)d_out;

    const int N = in_sizes[0] / (3 * 32 * 32);   // 512 images
    deq_kernel<<<dim3(N), dim3(256), LDS_FLOATS * sizeof(float), stream>>>(
        image, w1, b1, w2, b2, wh, bh, out);
}